// Gumbel_AP_GCN_72980084293672
// MI455X (gfx1250) — compile-verified
//
#include <hip/hip_runtime.h>
#include <hip/hip_bf16.h>
#include <math.h>

typedef float v2f __attribute__((ext_vector_type(2)));
typedef float v8f __attribute__((ext_vector_type(8)));

#define F_IN 512
#define HID  256
#define NCLS 64
#define NITER 10
#define LDSP (HID + 4)   // padded LDS row stride to dodge bank conflicts

// ---------------------------------------------------------------------------
// Fused MLP: out = relu(x @ W1 + b1) @ W2 + b2  using V_WMMA_F32_16X16X4_F32.
// Block = 256 threads (8 waves), handles 16 rows (nodes).
//   Stage 1: wave w computes h-tile cols [16w, 16w+16), K=512 loop -> LDS.
//   Stage 2: waves 0..3 compute out cols [16w, 16w+16), K=256 from LDS.
// ---------------------------------------------------------------------------
__global__ __launch_bounds__(256)
void fused_mlp_kernel(const float* __restrict__ x, const float* __restrict__ W1,
                      const float* __restrict__ b1, const float* __restrict__ W2,
                      const float* __restrict__ b2, float* __restrict__ ch,
                      float* __restrict__ pred0, int n) {
  __shared__ float hs[16 * LDSP];
  const int wave  = threadIdx.x >> 5;
  const int lane  = threadIdx.x & 31;
  const int lhalf = lane >> 4;      // 0: K={0,1}, 1: K={2,3}
  const int l16   = lane & 15;
  const long m0   = (long)blockIdx.x * 16;

  // ---- stage 1: h = relu(x @ W1 + b1) into LDS ----
  {
    const int  n0 = wave * 16;
    long m = m0 + l16; if (m > n - 1) m = n - 1;   // clamp tail reads
    const float* xrow = x + m * F_IN;
    v8f acc = {};
    for (int k0 = 0; k0 < F_IN; k0 += 4) {
      const int kk = k0 + lhalf * 2;
      v2f a, b;
      a.x = xrow[kk];
      a.y = xrow[kk + 1];
      b.x = W1[(long)kk       * HID + n0 + l16];
      b.y = W1[(long)(kk + 1) * HID + n0 + l16];
      acc = __builtin_amdgcn_wmma_f32_16x16x4_f32(false, a, false, b,
                                                  (short)0, acc, false, false);
    }
    const float bias = b1[n0 + l16];
#pragma unroll
    for (int v = 0; v < 8; ++v) {
      float hv = acc[v] + bias;
      hv = hv > 0.f ? hv : 0.f;                          // relu
      hs[(v + 8 * lhalf) * LDSP + n0 + l16] = hv;
    }
  }
  __syncthreads();

  // ---- stage 2: out = h @ W2 + b2 ----
  if (wave < 4) {
    const int n0 = wave * 16;
    const float* hrow = hs + (long)l16 * LDSP;
    v8f acc = {};
    for (int k0 = 0; k0 < HID; k0 += 4) {
      const int kk = k0 + lhalf * 2;
      v2f a, b;
      a.x = hrow[kk];                                    // ds_load
      a.y = hrow[kk + 1];
      b.x = W2[kk       * NCLS + n0 + l16];
      b.y = W2[(kk + 1) * NCLS + n0 + l16];
      acc = __builtin_amdgcn_wmma_f32_16x16x4_f32(false, a, false, b,
                                                  (short)0, acc, false, false);
    }
    const float bias = b2[n0 + l16];
#pragma unroll
    for (int v = 0; v < 8; ++v) {
      const long row = m0 + v + 8 * lhalf;
      if (row < n) {
        const float val = acc[v] + bias;
        ch[row * NCLS + n0 + l16]    = val;              // propagation input
        pred0[row * NCLS + n0 + l16] = val;              // predictions[0]
      }
    }
  }
}

// ---------------------------------------------------------------------------
// gcn_norm pieces
// ---------------------------------------------------------------------------
__global__ void deg_init_kernel(float* __restrict__ deg, int n) {
  int i = blockIdx.x * blockDim.x + threadIdx.x;
  if (i < n) deg[i] = 1.0f;                              // self loop
}

__global__ void deg_edge_kernel(const int* __restrict__ col, float* __restrict__ deg, int E) {
  int e = blockIdx.x * blockDim.x + threadIdx.x;
  if (e < E) atomicAdd(&deg[col[e]], 1.0f);
}

__global__ void dinv_kernel(const float* __restrict__ deg, float* __restrict__ dinv, int n) {
  int i = blockIdx.x * blockDim.x + threadIdx.x;
  if (i < n) {
    float d = deg[i];
    dinv[i] = d > 0.f ? rsqrtf(d) : 0.f;
  }
}

__global__ void norm_kernel(const int* __restrict__ row, const int* __restrict__ col,
                            const float* __restrict__ dinv, float* __restrict__ nrm, int E) {
  int e = blockIdx.x * blockDim.x + threadIdx.x;
  if (e < E) nrm[e] = dinv[row[e]] * dinv[col[e]];
}

__global__ void zero_kernel(float4* __restrict__ p, long n4) {
  long i = (long)blockIdx.x * blockDim.x + threadIdx.x;
  if (i < n4) p[i] = make_float4(0.f, 0.f, 0.f, 0.f);
}

// ---------------------------------------------------------------------------
// Scatter-add: 16 threads/edge, 4 floats/thread. ch/nh are L2-resident (25.6MB).
// ---------------------------------------------------------------------------
__global__ __launch_bounds__(256)
void scatter_kernel(const int* __restrict__ row, const int* __restrict__ col,
                    const float* __restrict__ nrm, const float* __restrict__ ch,
                    float* __restrict__ nh, int E) {
  long t = (long)blockIdx.x * blockDim.x + threadIdx.x;
  long e = t >> 4;
  if (e >= E) return;
  const int c4 = (int)(t & 15) * 4;
  const int r    = row[e];
  const int cdst = col[e];
  const float w  = nrm[e];
  const float4 v = *(const float4*)(ch + (long)r * NCLS + c4);
  float* dst = nh + (long)cdst * NCLS + c4;
  atomicAdd(dst + 0, v.x * w);
  atomicAdd(dst + 1, v.y * w);
  atomicAdd(dst + 2, v.z * w);
  atomicAdd(dst + 3, v.w * w);
}

// ---------------------------------------------------------------------------
// Per node: add self-loop msg, emit predictions slice, halt logit -> lambda.
// Block = 64 threads (2 waves) = one node, one feature per thread.
// ---------------------------------------------------------------------------
__global__ __launch_bounds__(64)
void finalize_kernel(const float* __restrict__ ch, float* __restrict__ nh,
                     const float* __restrict__ dinv, const float* __restrict__ halt_w,
                     const float* __restrict__ halt_b, float* __restrict__ pred,
                     float* __restrict__ lam, int n) {
  const long i = blockIdx.x;
  const int  c = threadIdx.x;
  const float di = dinv[i];
  const float sn = di * di;                              // self-loop norm
  float v = nh[i * NCLS + c] + ch[i * NCLS + c] * sn;
  nh[i * NCLS + c]   = v;                                // becomes next ch
  pred[i * NCLS + c] = v;
  float partial = v * halt_w[c];
  __shared__ float red[2];
  for (int off = 16; off > 0; off >>= 1)
    partial += __shfl_down(partial, off, 32);
  if ((c & 31) == 0) red[c >> 5] = partial;
  __syncthreads();
  if (c == 0) {
    const float logit = red[0] + red[1] + halt_b[0];
    lam[i] = 1.0f / (1.0f + __expf(-logit));
  }
}

// ---------------------------------------------------------------------------
// Halting probabilities: p[t] = lam[t]*prod_{s<t}(1-lam[s]); leftover -> last.
// ---------------------------------------------------------------------------
__global__ void halt_prob_kernel(const float* __restrict__ lam, float* __restrict__ p, int n) {
  int i = blockIdx.x * blockDim.x + threadIdx.x;
  if (i >= n) return;
  float q = 1.f;
  float pv[NITER];
#pragma unroll
  for (int t = 0; t < NITER; ++t) {
    float l = lam[(long)t * n + i];
    pv[t] = l * q;
    q *= (1.f - l);
  }
  pv[NITER - 1] += q;
#pragma unroll
  for (int t = 0; t < NITER; ++t) p[(long)i * NITER + t] = pv[t];
}

// ---------------------------------------------------------------------------
extern "C" void kernel_launch(void* const* d_in, const int* in_sizes, int n_in,
                              void* d_out, int out_size, void* d_ws, size_t ws_size,
                              hipStream_t stream) {
  const float* x   = (const float*)d_in[0];
  const int*   ei  = (const int*)d_in[1];
  // d_in[2] edge_dropout: identity in eval mode, ignored.
  const float* W1  = (const float*)d_in[3];
  const float* b1  = (const float*)d_in[4];
  const float* W2  = (const float*)d_in[5];
  const float* b2  = (const float*)d_in[6];
  const float* hw  = (const float*)d_in[7];
  const float* hb  = (const float*)d_in[8];

  const int n = in_sizes[0] / F_IN;    // 100000
  const int E = in_sizes[1] / 2;       // 3200000
  const int* row  = ei;                // edge_index[0] (source)
  const int* colv = ei + E;            // edge_index[1] (target)
  float* out = (float*)d_out;

  // workspace carve-out
  char* wsp = (char*)d_ws;
  auto carve = [&](size_t bytes) -> void* {
    void* p = (void*)wsp;
    wsp += (bytes + 255) & ~(size_t)255;
    return p;
  };
  float* chA  = (float*)carve((size_t)n * NCLS * sizeof(float));
  float* chB  = (float*)carve((size_t)n * NCLS * sizeof(float));
  float* deg  = (float*)carve((size_t)n * sizeof(float));
  float* dinv = (float*)carve((size_t)n * sizeof(float));
  float* nrm  = (float*)carve((size_t)E * sizeof(float));
  float* lam  = (float*)carve((size_t)NITER * (size_t)n * sizeof(float));

  // MLP (WMMA) -> chA and predictions[0]
  fused_mlp_kernel<<<(n + 15) / 16, 256, 0, stream>>>(x, W1, b1, W2, b2, chA, out, n);

  // gcn_norm
  deg_init_kernel<<<(n + 255) / 256, 256, 0, stream>>>(deg, n);
  deg_edge_kernel<<<(E + 255) / 256, 256, 0, stream>>>(colv, deg, E);
  dinv_kernel<<<(n + 255) / 256, 256, 0, stream>>>(deg, dinv, n);
  norm_kernel<<<(E + 255) / 256, 256, 0, stream>>>(row, colv, dinv, nrm, E);

  // 10 propagation steps
  float* ch = chA;
  float* nh = chB;
  const long n4   = (long)n * NCLS / 4;
  const long scat = (long)E * 16;
  for (int t = 0; t < NITER; ++t) {
    zero_kernel<<<(unsigned)((n4 + 255) / 256), 256, 0, stream>>>((float4*)nh, n4);
    scatter_kernel<<<(unsigned)((scat + 255) / 256), 256, 0, stream>>>(row, colv, nrm, ch, nh, E);
    finalize_kernel<<<n, 64, 0, stream>>>(ch, nh, dinv, hw, hb,
                                          out + (long)(t + 1) * n * NCLS,
                                          lam + (long)t * n, n);
    float* tmp = ch; ch = nh; nh = tmp;
  }

  // p [n, NITER] after predictions [(NITER+1), n, NCLS]
  halt_prob_kernel<<<(n + 255) / 256, 256, 0, stream>>>(
      lam, out + (long)(NITER + 1) * (long)n * NCLS, n);
}